// RadarPts_73074573574698
// MI455X (gfx1250) — compile-verified
//
#include <hip/hip_runtime.h>
#include <stdint.h>

#define BATCH 8
#define NCLS 10
#define H 512
#define W 512
#define HW (H * W)
#define BAND 8           // rows computed per band
#define BROWS (BAND + 2) // +2 halo rows
#define GROUPS 8         // band-groups per plane (each group = 64 rows)
#define NCHUNK 1024      // HW / 256 chunks per plane
#define CHUNKSZ 256      // pixels per chunk (half a row)

// ---------- CDNA5 async global->LDS copy helpers (ASYNCcnt path) ----------
__device__ __forceinline__ uint32_t lds_off32(const void* p) {
  // low 32 bits of a generic LDS pointer == byte offset within the wave's
  // LDS allocation (ISA 10.2).
  return (uint32_t)(uintptr_t)p;
}
__device__ __forceinline__ void async_ld_b128(uint32_t ldsoff, const float* g) {
  asm volatile("global_load_async_to_lds_b128 %0, %1, off"
               :: "v"(ldsoff), "v"(g) : "memory");
}
__device__ __forceinline__ void async_ld_b32(uint32_t ldsoff, const float* g) {
  asm volatile("global_load_async_to_lds_b32 %0, %1, off"
               :: "v"(ldsoff), "v"(g) : "memory");
}
__device__ __forceinline__ void wait_async_le2() {
  asm volatile("s_wait_asynccnt 2" ::: "memory");
}
__device__ __forceinline__ void wait_async_0() {
  asm volatile("s_wait_asynccnt 0" ::: "memory");
}

// pack (value, pixel-index): unsigned max => larger value wins, ties -> smaller
// index wins (matches jax.lax.top_k tie order). Valid because suppressed >= 0.
__device__ __forceinline__ unsigned long long packv(float v, uint32_t pix) {
  return ((unsigned long long)__float_as_uint(v) << 32) |
         (unsigned long long)(0xFFFFFFFFu - pix);
}

// wave32 max-reduction: 5 shuffle steps, no barriers
__device__ __forceinline__ unsigned long long wave_max(unsigned long long v) {
#pragma unroll
  for (int m = 16; m > 0; m >>= 1) {
    unsigned long long o = __shfl_xor(v, m, 32);
    if (o > v) v = o;
  }
  return v;
}

__device__ __forceinline__ float suppress_global(const float* __restrict__ plane,
                                                 int gy, int gx) {
  float c = plane[gy * W + gx];
  float m = c;
#pragma unroll
  for (int dy = -1; dy <= 1; ++dy) {
    int yy = gy + dy;
    if (yy < 0 || yy >= H) continue;
#pragma unroll
    for (int dx = -1; dx <= 1; ++dx) {
      int xx = gx + dx;
      if (xx < 0 || xx >= W) continue;
      m = fmaxf(m, plane[yy * W + xx]);
    }
  }
  return (m == c) ? c : 0.0f;
}

// ---------------- Stage 1a: NMS + per-chunk packed maxima ----------------
// grid: (GROUPS, BATCH*NCLS), block: 1024
__global__ __launch_bounds__(1024) void ctdet_nms_chunkmax(
    const float* __restrict__ heat, unsigned long long* __restrict__ chunkws) {
  __shared__ __align__(16) float band[2][BROWS * W];   // 40 KB, double buffered
  __shared__ unsigned long long chunkL[2 * 64];        // 128 chunks per block

  const int tid = threadIdx.x;
  const int g = blockIdx.x;       // band group: rows [g*64, g*64+64)
  const int plane = blockIdx.y;   // b*NCLS + c
  const float* hp = heat + (size_t)plane * HW;

  if (tid < 128) chunkL[tid] = 0ull;

  // 2 async instructions per wave per band: one b128 (4 dwords) + one b32
  auto issue = [&](int bi, int buf) {
    const int r0 = g * 64 + bi * BAND;
    {
      int lin = tid * 4;                         // 0..4095 (16B aligned)
      int row = lin >> 9;
      int col = lin & (W - 1);
      int grow = r0 - 1 + row;
      grow = grow < 0 ? 0 : (grow >= H ? H - 1 : grow);
      async_ld_b128(lds_off32(&band[buf][row * W + col]), hp + grow * W + col);
    }
    {
      int lin = 4096 + tid;                      // 4096..5119 remainder
      int row = lin >> 9;
      int col = lin & (W - 1);
      int grow = r0 - 1 + row;
      grow = grow < 0 ? 0 : (grow >= H ? H - 1 : grow);
      async_ld_b32(lds_off32(&band[buf][row * W + col]), hp + grow * W + col);
    }
  };

  issue(0, 0);
  for (int bi = 0; bi < 64 / BAND; ++bi) {
    const int cur = bi & 1;
    // async loads complete in order: after issuing the next band's 2 ops,
    // cnt<=2 means the current band's 2 ops have landed.
    if (bi + 1 < 64 / BAND) { issue(bi + 1, 1 - cur); wait_async_le2(); }
    else                    { wait_async_0(); }
    __syncthreads();   // all waves' band-bi async writes visible

    const int r0 = g * 64 + bi * BAND;
    const int lp = tid * 4;            // 4 consecutive pixels, same chunk
    const int rb = lp >> 9;            // row within band 0..7
    const int gx0 = lp & (W - 1);
    const int gy = r0 + rb;
    unsigned long long best = 0ull;
#pragma unroll
    for (int j = 0; j < 4; ++j) {
      const int gx = gx0 + j;
      float c = band[cur][(rb + 1) * W + gx];
      float m = c;
      for (int dy = -1; dy <= 1; ++dy) {
        int yy = gy + dy;
        if (yy < 0 || yy >= H) continue;
        for (int dx = -1; dx <= 1; ++dx) {
          int xx = gx + dx;
          if (xx < 0 || xx >= W) continue;
          m = fmaxf(m, band[cur][(yy - r0 + 1) * W + xx]);
        }
      }
      float s = (m == c) ? c : 0.0f;
      unsigned long long pk = packv(s, (uint32_t)(gy * W + gx));
      best = best > pk ? best : pk;
    }
    // local chunk id = bi*16 + tid/64 (exclusive to this block)
    atomicMax(&chunkL[bi * 16 + (tid >> 6)], best);   // ds_max_u64
    __syncthreads();
  }
  if (tid < 128)
    chunkws[(size_t)plane * NCHUNK + g * 128 + tid] = chunkL[tid];
}

// ---------------- Stage 1b: per-class exact top-K by iterative selection ----
// grid: BATCH*NCLS, block: 1024
__global__ __launch_bounds__(1024) void ctdet_topk_perclass(
    const float* __restrict__ heat, const unsigned long long* __restrict__ chunkws,
    const int* __restrict__ Kp, float* __restrict__ sc1, uint32_t* __restrict__ id1) {
  __shared__ unsigned long long chunkpack[NCHUNK];
  __shared__ unsigned long long part[32];
  __shared__ uint32_t sel[128];

  const int tid = threadIdx.x;
  const int lane = tid & 31;
  const int wid = tid >> 5;
  const int plane = blockIdx.x;
  const int K = *Kp;
  const float* hp = heat + (size_t)plane * HW;

  chunkpack[tid] = chunkws[(size_t)plane * NCHUNK + tid];
  __syncthreads();

  for (int it = 0; it < K; ++it) {
    // --- argmax over 1024 chunk maxima: shuffle reduce + 32-partial combine
    unsigned long long v = wave_max(chunkpack[tid]);
    if (lane == 0) part[wid] = v;
    __syncthreads();
    if (tid < 32) {
      unsigned long long w = wave_max(part[tid]);
      if (tid == 0) part[0] = w;
    }
    __syncthreads();
    const unsigned long long p = part[0];

    uint32_t pix = 0xFFFFFFFFu - (uint32_t)(p & 0xFFFFFFFFull);
    float val = __uint_as_float((uint32_t)(p >> 32));
    if (p == 0ull) { pix = 0; val = 0.0f; }
    if (tid == 0) {
      sc1[plane * K + it] = val;
      id1[plane * K + it] = pix;
      sel[it] = (p == 0ull) ? 0xFFFFFFFFu : pix;
    }
    __syncthreads();

    if (p != 0ull) {               // uniform branch (p from LDS)
      const uint32_t ch = pix >> 8;
      unsigned long long np = 0ull;
      if (tid < CHUNKSZ) {
        const uint32_t q = ch * CHUNKSZ + tid;
        bool removed = false;
        for (int t = 0; t <= it; ++t) removed |= (sel[t] == q);
        if (!removed) np = packv(suppress_global(hp, q >> 9, q & (W - 1)), q);
      }
      np = wave_max(np);
      if (lane == 0) part[wid] = np;
      __syncthreads();
      if (tid < 32) {
        unsigned long long w = wave_max(part[tid]);
        if (tid == 0) chunkpack[ch] = w;
      }
    }
    __syncthreads();
  }
}

// ---------------- Stage 2: global top-K over C*K candidates per batch -------
// grid: BATCH, block: 1024
__global__ __launch_bounds__(1024) void ctdet_topk_global(
    const float* __restrict__ sc1, const uint32_t* __restrict__ id1,
    const int* __restrict__ Kp, float* __restrict__ out) {
  __shared__ unsigned long long cand[1024];
  __shared__ unsigned long long part[32];

  const int tid = threadIdx.x;
  const int lane = tid & 31;
  const int wid = tid >> 5;
  const int b = blockIdx.x;
  const int K = *Kp;
  const int CK = NCLS * K;     // 1000
  const int BK = BATCH * K;    // 800

  unsigned long long v = 0ull;
  if (tid < CK)
    v = ((unsigned long long)__float_as_uint(sc1[b * CK + tid]) << 32) |
        (unsigned long long)(0xFFFFFFFFu - (uint32_t)tid);
  cand[tid] = v;
  __syncthreads();

  for (int it = 0; it < K; ++it) {
    unsigned long long r = wave_max(cand[tid]);
    if (lane == 0) part[wid] = r;
    __syncthreads();
    if (tid < 32) {
      unsigned long long w = wave_max(part[tid]);
      if (tid == 0) part[0] = w;
    }
    __syncthreads();
    if (tid == 0) {
      const unsigned long long p = part[0];
      const uint32_t j = 0xFFFFFFFFu - (uint32_t)(p & 0xFFFFFFFFull);
      float val = __uint_as_float((uint32_t)(p >> 32));
      uint32_t ind = 0, cls = 0;
      if (p != 0ull && j < (uint32_t)CK) {
        ind = id1[b * CK + j];
        cls = j / (uint32_t)K;
        cand[j] = 0ull;                 // remove selected candidate
      } else {
        val = 0.0f;
      }
      out[         b * K + it] = val;                    // topk_score
      out[BK     + b * K + it] = (float)ind;             // topk_inds
      out[2 * BK + b * K + it] = (float)cls;             // topk_clses
      out[3 * BK + b * K + it] = (float)(ind >> 9);      // topk_ys = ind / 512
      out[4 * BK + b * K + it] = (float)(ind & (W - 1)); // topk_xs = ind % 512
    }
    __syncthreads();
  }
}

extern "C" void kernel_launch(void* const* d_in, const int* in_sizes, int n_in,
                              void* d_out, int out_size, void* d_ws, size_t ws_size,
                              hipStream_t stream) {
  (void)in_sizes; (void)n_in; (void)out_size; (void)ws_size;
  const float* heat = (const float*)d_in[0];
  const int* Kp = (const int*)d_in[1];

  // workspace layout: chunk maxima (80*1024 u64 = 640 KB), then sc1/id1
  unsigned long long* chunkws = (unsigned long long*)d_ws;
  char* p1 = (char*)d_ws + (size_t)BATCH * NCLS * NCHUNK * sizeof(unsigned long long);
  float* sc1 = (float*)p1;                                   // [80][K<=128]
  uint32_t* id1 = (uint32_t*)(p1 + BATCH * NCLS * 128 * sizeof(float));

  dim3 gridA(GROUPS, BATCH * NCLS);
  ctdet_nms_chunkmax<<<gridA, 1024, 0, stream>>>(heat, chunkws);
  ctdet_topk_perclass<<<BATCH * NCLS, 1024, 0, stream>>>(heat, chunkws, Kp, sc1, id1);
  ctdet_topk_global<<<BATCH, 1024, 0, stream>>>(sc1, id1, Kp, (float*)d_out);
}